// ManifoldMHA_1906965479917
// MI455X (gfx1250) — compile-verified
//
#include <hip/hip_runtime.h>
#include <hip/hip_bf16.h>
#include <math.h>

// Problem constants (from reference): S=256, B=2, HID=512, NH=8, HD=64, C=1
#define S_    256
#define B_    2
#define HID_  512
#define NH_   8
#define HD_   64
#define EPSF  1e-15f

typedef __attribute__((ext_vector_type(2))) float v2f;
typedef __attribute__((ext_vector_type(8))) float v8f;

// ---------------------------------------------------------------- helpers ---

__device__ __forceinline__ float artanh_c(float z) {
  z = fminf(fmaxf(z, -1.0f + 1e-7f), 1.0f - 1e-7f);
  return 0.5f * logf((1.0f + z) / (1.0f - z));
}

// D = A(16x4 f32) * B(4x16 f32) + C  -- v_wmma_f32_16x16x4_f32
__device__ __forceinline__ v8f wmma_f32x4(v2f a, v2f b, v8f c) {
  return __builtin_amdgcn_wmma_f32_16x16x4_f32(
      /*neg_a=*/false, a, /*neg_b=*/false, b,
      /*c_mod=*/(short)0, c, /*reuse_a=*/false, /*reuse_b=*/false);
}

// power-of-2 block sum reduction, result broadcast to all threads
__device__ __forceinline__ float block_sum(float v, float* red, int n) {
  int tid = threadIdx.x;
  red[tid] = v;
  __syncthreads();
  for (int off = n >> 1; off > 0; off >>= 1) {
    if (tid < off) red[tid] += red[tid + off];
    __syncthreads();
  }
  float r = red[0];
  __syncthreads();
  return r;
}

// ------------------------------------------------- kernel 1: mx = x @ W^T ---
// x[row=b*S+s][h] = query[s][b][h]; one wave per 16x16 tile, K=512 in steps of 4.
__global__ void k_proj_gemm(const float* __restrict__ query,
                            const float* __restrict__ Wq,
                            const float* __restrict__ Wk,
                            const float* __restrict__ Wv,
                            float* __restrict__ mx /* [3][512][512] */) {
  int wave = (blockIdx.x * blockDim.x + threadIdx.x) >> 5;
  int lane = threadIdx.x & 31;
  int mat  = wave >> 10;          // 1024 tiles per matrix
  int tile = wave & 1023;
  int tr   = tile >> 5;           // 0..31 row tile
  int tc   = tile & 31;           // 0..31 col tile
  const float* W = (mat == 0) ? Wq : ((mat == 1) ? Wk : Wv);

  int half = lane >> 4, l = lane & 15;
  int arow = tr * 16 + l;                      // row of x (A matrix, M = l)
  int b = arow / S_, s = arow % S_;
  const float* xrow = query + ((size_t)s * B_ + b) * HID_;
  const float* wrow = W + (size_t)(tc * 16 + l) * HID_;   // B[k][n=l] = W[o=n][k]

  v8f c = {};
  for (int kk = 0; kk < HID_; kk += 4) {
    int k0 = kk + half * 2;
    v2f a  = { xrow[k0], xrow[k0 + 1] };
    v2f bb = { wrow[k0], wrow[k0 + 1] };
    c = wmma_f32x4(a, bb, c);
  }
  float* out = mx + (size_t)mat * HID_ * HID_;
  for (int r = 0; r < 8; ++r) {
    int m = r + 8 * half;
    out[(size_t)(tr * 16 + m) * HID_ + (tc * 16 + l)] = c[r];
  }
}

// --------------------------- kernel 2: row-wise manifold linear + log/exp ---
// One 256-thread block per (mat, b, s). Produces ql/kl/vl rows [b][r=s*8+h][d]
// plus q2/k2/gamma stats.
__global__ void k_manifold_rows(const float* __restrict__ query,
                                const float* __restrict__ bq,
                                const float* __restrict__ bk,
                                const float* __restrict__ bv,
                                const float* __restrict__ mx,
                                float* __restrict__ qkl,   // [3][B][2048][64]
                                float* __restrict__ q2,    // [B*2048]
                                float* __restrict__ k2,
                                float* __restrict__ gam) {
  __shared__ float red[256];
  __shared__ float chunk[8];
  int tid = threadIdx.x;
  int mat = blockIdx.x / (B_ * S_);
  int rem = blockIdx.x % (B_ * S_);
  int b = rem >> 8, s = rem & 255;
  const float* bias = (mat == 0) ? bq : ((mat == 1) ? bk : bv);

  int e0 = tid, e1 = tid + 256;
  const float* mrow = mx + ((size_t)mat * (B_ * S_) + (size_t)b * S_ + s) * HID_;
  const float* xrow = query + ((size_t)s * B_ + b) * HID_;
  float m0 = mrow[e0], m1 = mrow[e1];
  float x0 = xrow[e0], x1 = xrow[e1];
  float bb0 = bias[e0], bb1 = bias[e1];

  float sum_m2 = block_sum(m0 * m0 + m1 * m1, red, 256);
  float sum_x2 = block_sum(x0 * x0 + x1 * x1, red, 256);
  float nz     = block_sum((m0 != 0.0f || m1 != 0.0f) ? 1.0f : 0.0f, red, 256);

  // mobius_matvec tail
  float mxn  = fmaxf(sqrtf(sum_m2), EPSF);
  float xn   = fmaxf(sqrtf(sum_x2), EPSF);
  float resn = tanhf(mxn / xn * artanh_c(xn));
  float sc   = (nz > 0.0f) ? (resn / mxn) : 0.0f;
  float r0 = m0 * sc, r1 = m1 * sc;
  float x2v = (nz > 0.0f) ? resn * resn : 0.0f;

  // mobius_add(res, bias) ; C = 1
  float y2 = block_sum(bb0 * bb0 + bb1 * bb1, red, 256);
  float xy = block_sum(r0 * bb0 + r1 * bb1, red, 256);
  float Ac = 1.0f + 2.0f * xy + y2;
  float Bc = 1.0f - x2v;
  float den = fmaxf(1.0f + 2.0f * xy + x2v * y2, EPSF);
  float w0 = (Ac * r0 + Bc * bb0) / den;
  float w1 = (Ac * r1 + Bc * bb1) / den;

  // projx
  float sw2 = block_sum(w0 * w0 + w1 * w1, red, 256);
  float n = fmaxf(sqrtf(sw2), EPSF);
  const float maxn = 1.0f - 4e-3f;
  if (n > maxn) {
    float f = maxn / n;
    w0 *= f; w1 *= f;
    n = maxn;
  }
  // logmap0 over 512
  float lu = artanh_c(n) / n;
  float u0 = w0 * lu, u1 = w1 * lu;

  // expmap0 per 64-chunk
  if (tid < 8) chunk[tid] = 0.0f;
  __syncthreads();
  atomicAdd(&chunk[e0 >> 6], u0 * u0);
  atomicAdd(&chunk[e1 >> 6], u1 * u1);
  __syncthreads();

  float cn0 = fmaxf(sqrtf(chunk[e0 >> 6]), EPSF);
  float cn1 = fmaxf(sqrtf(chunk[e1 >> 6]), EPSF);
  float o0 = u0 * (tanhf(cn0) / cn0);
  float o1 = u1 * (tanhf(cn1) / cn1);

  float* dst = qkl + ((size_t)mat * B_ + b) * 2048 * 64;
  int rr0 = s * 8 + (e0 >> 6), rr1 = s * 8 + (e1 >> 6);
  dst[(size_t)rr0 * 64 + (e0 & 63)] = o0;
  dst[(size_t)rr1 * 64 + (e1 & 63)] = o1;

  if (tid < 8) {
    float t = tanhf(fmaxf(sqrtf(chunk[tid]), EPSF));   // row norm of output chunk
    int idx = b * 2048 + s * 8 + tid;
    if (mat == 0)      q2[idx]  = t * t;
    else if (mat == 1) k2[idx]  = t * t;
    else               gam[idx] = 2.0f / fmaxf(1.0f - t * t, EPSF);
  }
}

// ------------------------------- kernel 3: hyperbolic scores + sigmoid ------
// One wave per 16x16 (i,j) tile of one (b,h). WMMA for q.k dots, then
// per-pair reciprocal-norm pass from LDS.
__global__ void k_scores(const float* __restrict__ ql,   // [B][2048][64]
                         const float* __restrict__ kl,
                         const float* __restrict__ q2,
                         const float* __restrict__ k2,
                         float* __restrict__ probs) {    // [B][H][S][S]
  __shared__ float sq[16][64];
  __shared__ float sk[16][64];
  int lane = threadIdx.x & 31;
  int bi = blockIdx.x;
  int t  = bi & 255;          // 16x16 tile grid over (i,j)
  int h  = (bi >> 8) & 7;
  int b  = bi >> 11;
  int iT = (t >> 4) * 16, jT = (t & 15) * 16;

  const float* qbase = ql + ((size_t)b * 2048 + h * 256) * 64;
  const float* kbase = kl + ((size_t)b * 2048 + h * 256) * 64;
  for (int it = 0; it < 32; ++it) {
    int lin = it * 32 + lane;             // 1024 elements each
    int i = lin >> 6, d = lin & 63;
    sq[i][d] = qbase[(size_t)(iT + i) * 64 + d];
    sk[i][d] = kbase[(size_t)(jT + i) * 64 + d];
  }
  __syncthreads();

  int half = lane >> 4, l = lane & 15;
  v8f c = {};
  for (int kk = 0; kk < 64; kk += 4) {
    int k0 = kk + half * 2;
    v2f a  = { sq[l][k0], sq[l][k0 + 1] };
    v2f bb = { sk[l][k0], sk[l][k0 + 1] };
    c = wmma_f32x4(a, bb, c);
  }

  for (int r = 0; r < 8; ++r) {
    int i = r + 8 * half;   // local i
    int j = l;              // local j
    float dot = c[r];
    float x2 = q2[b * 2048 + h * 256 + iT + i];
    float y2 = k2[b * 2048 + h * 256 + jT + j];
    float xy = -dot;                                // x = -q
    float Ac = 1.0f + 2.0f * xy + y2;
    float Bc = 1.0f - x2;
    float invden = 1.0f / fmaxf(1.0f + 2.0f * xy + x2 * y2, EPSF);
    float ssum = 0.0f;
    for (int d = 0; d < 64; ++d) {
      float num = Bc * sk[j][d] - Ac * sq[i][d];    // A*(-q) + B*k
      float df = num * invden;
      float d2 = fmaxf(df * df, 1e-30f);
      ssum += 1.0f / d2;
    }
    float t1 = 1.0f / sqrtf(ssum);
    float score = -2.0f * artanh_c(t1);
    score = fminf(fmaxf(score, -1e10f), 1e10f);
    float p = 1.0f / (1.0f + expf(-score));
    probs[(((size_t)b * NH_ + h) * S_ + (iT + i)) * S_ + (jT + j)] = p;
  }
}

// -------------------------- kernel 4a: denom = probs @ (gamma - 1) ----------
__global__ void k_denom(const float* __restrict__ probs,
                        const float* __restrict__ gam,
                        float* __restrict__ denom) {   // [B*H*S]
  int q = threadIdx.x;
  int h = blockIdx.x & 7;
  int b = blockIdx.x >> 3;
  const float* P = probs + (((size_t)b * NH_ + h) * S_ + q) * S_;
  const float* G = gam + b * 2048 + h * 256;
  float acc = 0.0f;
  for (int n = 0; n < S_; ++n) acc += P[n] * (G[n] - 1.0f);
  float sgn = (acc >= 0.0f) ? 1.0f : -1.0f;
  denom[((size_t)b * NH_ + h) * S_ + q] = sgn * fmaxf(fabsf(acc), 1e-10f);
}

// -------------------------- kernel 4b: nom = probs @ (gamma * vl) via WMMA --
// One wave per (b,h,qT,dT) tile; K=256 over n.
__global__ void k_nom(const float* __restrict__ probs,
                      const float* __restrict__ gam,
                      const float* __restrict__ vl,     // [B][2048][64]
                      float* __restrict__ nom) {        // [B][H][S][64]
  int lane = threadIdx.x & 31;
  int bi = blockIdx.x;
  int dT = bi & 3;
  int qT = ((bi >> 2) & 15) * 16;
  int h  = (bi >> 6) & 7;
  int b  = bi >> 9;

  int half = lane >> 4, l = lane & 15;
  const float* P = probs + (((size_t)b * NH_ + h) * S_) * S_;
  const float* V = vl + ((size_t)b * 2048 + h * 256) * 64;
  const float* G = gam + b * 2048 + h * 256;

  v8f c = {};
  for (int kk = 0; kk < S_; kk += 4) {
    int k0 = kk + half * 2;
    v2f a  = { P[(size_t)(qT + l) * S_ + k0], P[(size_t)(qT + l) * S_ + k0 + 1] };
    float g0 = G[k0], g1 = G[k0 + 1];
    v2f bb = { g0 * V[(size_t)k0 * 64 + dT * 16 + l],
               g1 * V[(size_t)(k0 + 1) * 64 + dT * 16 + l] };
    c = wmma_f32x4(a, bb, c);
  }
  float* out = nom + (((size_t)b * NH_ + h) * S_) * 64;
  for (int r = 0; r < 8; ++r) {
    int m = r + 8 * half;
    out[(size_t)(qT + m) * 64 + dT * 16 + l] = c[r];
  }
}

// ---------------- kernel 5: ctx = logmap0(mobius_scalar_mul(0.5, nom/den)) --
__global__ void k_ctxlog(const float* __restrict__ nom,
                         const float* __restrict__ denom,
                         float* __restrict__ ctxlog) {  // [B][H][S][64]
  __shared__ float red[64];
  int d = threadIdx.x;
  int q = blockIdx.x & 255;
  int h = (blockIdx.x >> 8) & 7;
  int b = blockIdx.x >> 11;
  size_t row = ((size_t)b * NH_ + h) * S_ + q;
  float u = nom[row * 64 + d] / denom[row];
  float s2 = block_sum(u * u, red, 64);
  float n = fmaxf(sqrtf(s2), EPSF);
  float t = tanhf(0.5f * artanh_c(n));     // |mobius_scalar_mul result|
  float res = t * u / n;
  float nres = fmaxf(t, EPSF);
  ctxlog[row * 64 + d] = res * (artanh_c(nres) / nres);
}

// ---------------- kernel 6: expmap0 over 512 + transpose to [S,B,HID] -------
__global__ void k_final(const float* __restrict__ ctxlog,
                        float* __restrict__ out) {      // [S][B][HID]
  __shared__ float red[256];
  int tid = threadIdx.x;
  int s = blockIdx.x & 255;
  int b = blockIdx.x >> 8;
  int e0 = tid, e1 = tid + 256;
  // ctx_t[b,s,h,d] = ctxlog[b][h][s][d]
  float v0 = ctxlog[(((size_t)b * NH_ + (e0 >> 6)) * S_ + s) * 64 + (e0 & 63)];
  float v1 = ctxlog[(((size_t)b * NH_ + (e1 >> 6)) * S_ + s) * 64 + (e1 & 63)];
  float s2 = block_sum(v0 * v0 + v1 * v1, red, 256);
  float n = fmaxf(sqrtf(s2), EPSF);
  float f = tanhf(n) / n;
  out[((size_t)s * B_ + b) * HID_ + e0] = v0 * f;
  out[((size_t)s * B_ + b) * HID_ + e1] = v1 * f;
}

// --------------------------------------------------------------- launcher ---
extern "C" void kernel_launch(void* const* d_in, const int* in_sizes, int n_in,
                              void* d_out, int out_size, void* d_ws, size_t ws_size,
                              hipStream_t stream) {
  const float* query = (const float*)d_in[0];
  const float* Wq = (const float*)d_in[1];
  const float* bq = (const float*)d_in[2];
  const float* Wk = (const float*)d_in[3];
  const float* bk = (const float*)d_in[4];
  const float* Wv = (const float*)d_in[5];
  const float* bv = (const float*)d_in[6];
  float* out = (float*)d_out;
  float* ws = (float*)d_ws;

  // workspace layout (floats); mx region reused for nom/denom/ctxlog later
  const size_t MX_OFF   = 0;        // 3*512*512 = 786432 (kernels 1-2)
  const size_t NOM_OFF  = 0;        // 262144            (kernels 4b+)
  const size_t DEN_OFF  = 262144;   // 4096
  const size_t CLOG_OFF = 266240;   // 262144
  const size_t QKL_OFF  = 786432;   // 3*2*2048*64 = 786432
  const size_t Q2_OFF   = 1572864;  // 4096
  const size_t K2_OFF   = 1576960;  // 4096
  const size_t GAM_OFF  = 1581056;  // 4096  -> total 1585152 floats ~ 6.3 MB

  float* mx   = ws + MX_OFF;
  float* qkl  = ws + QKL_OFF;
  float* qlw  = qkl;
  float* klw  = qkl + 2 * 2048 * 64;
  float* vlw  = qkl + 4 * 2048 * 64;
  float* q2   = ws + Q2_OFF;
  float* k2   = ws + K2_OFF;
  float* gam  = ws + GAM_OFF;
  float* nomw = ws + NOM_OFF;
  float* denw = ws + DEN_OFF;
  float* clog = ws + CLOG_OFF;

  const size_t CTXN = (size_t)S_ * B_ * HID_;   // 262144
  float* probs = out + CTXN;                    // [B][H][S][S]

  // 1) three projection GEMMs: 3*1024 tiles, 8 waves/block
  k_proj_gemm<<<384, 256, 0, stream>>>(query, Wq, Wk, Wv, mx);
  // 2) row-wise manifold math: (3 mats) x (B*S rows)
  k_manifold_rows<<<3 * B_ * S_, 256, 0, stream>>>(query, bq, bk, bv, mx,
                                                   qkl, q2, k2, gam);
  // 3) scores + sigmoid: B*H*256 tiles, one wave each
  k_scores<<<B_ * NH_ * 256, 32, 0, stream>>>(qlw, klw, q2, k2, probs);
  // 4) gyro-midpoint
  k_denom<<<B_ * NH_, 256, 0, stream>>>(probs, gam, denw);
  k_nom<<<B_ * NH_ * 16 * 4, 32, 0, stream>>>(probs, gam, vlw, nomw);
  // 5) mobius half-scale + logmap0 (fused)
  k_ctxlog<<<B_ * NH_ * S_, 64, 0, stream>>>(nomw, denw, clog);
  // 6) expmap0 over HID + final transpose
  k_final<<<B_ * S_, 256, 0, stream>>>(clog, out);
}